// SAGE_17248588661531
// MI455X (gfx1250) — compile-verified
//
#include <hip/hip_runtime.h>
#include <hip/hip_bf16.h>

typedef float v2f __attribute__((ext_vector_type(2)));
typedef float v8f __attribute__((ext_vector_type(8)));

// ---------------------------------------------------------------------------
// Degree kernels (computed once; deg buffer turned into 1/max(deg,1) in place)
// ---------------------------------------------------------------------------
__global__ void sage_deg_kernel(const long long* __restrict__ dst,
                                float* __restrict__ deg, int E) {
    int e = blockIdx.x * blockDim.x + threadIdx.x;
    if (e < E) unsafeAtomicAdd(&deg[dst[e]], 1.0f);
}

__global__ void sage_rdeg_kernel(float* __restrict__ deg, int N) {
    int i = blockIdx.x * blockDim.x + threadIdx.x;
    if (i < N) deg[i] = 1.0f / fmaxf(deg[i], 1.0f);
}

// ---------------------------------------------------------------------------
// Edge scatter-add: one wave per edge, 4 features per lane (float4 gather,
// 4x global_atomic_add_f32 scatter — agg working set lives in the 192MB L2)
// ---------------------------------------------------------------------------
__global__ void sage_agg_kernel(const long long* __restrict__ src,
                                const long long* __restrict__ dst,
                                const float* __restrict__ feat,
                                float* __restrict__ agg, int E) {
    long long tid = (long long)blockIdx.x * blockDim.x + threadIdx.x;
    int e = (int)(tid >> 5);
    if (e >= E) return;
    int f = ((int)tid & 31) * 4;
    long long s = src[e];
    long long d = dst[e];
    const float4 v = *(const float4*)(feat + (size_t)s * 128 + f);
    float* ap = agg + (size_t)d * 128 + f;
    unsafeAtomicAdd(ap + 0, v.x);
    unsafeAtomicAdd(ap + 1, v.y);
    unsafeAtomicAdd(ap + 2, v.z);
    unsafeAtomicAdd(ap + 3, v.w);
}

// ---------------------------------------------------------------------------
// Fused SAGEConv GEMM:  out = (agg*rdeg) @ Wl^T + feat @ Wr^T + bl  [, relu]
// 256 threads = 8 waves; wave w owns output columns [16w,16w+16).
// Persistent blocks. Both weight matrices staged ONCE per block into LDS
// (CDNA5: 320KB/WGP; we use ~148.5KB -> 2 blocks/WGP), stride-132 padded so
// the 16 B-fragment rows land on banks 4 apart and the b64 reads of the two
// lane halves tile all 64 banks exactly once. No VGPR arrays -> no spills.
//
// V_WMMA_F32_16X16X4_F32 layout (ISA 7.12.2):
//   A 16x4:  lane<16 -> M=lane, K={0,1}; lane>=16 -> M=lane-16, K={2,3}
//   B 4x16:  mirrored (N=lane&15, same K split)
//   C/D:     reg r -> row r + 8*(lane>>4), col = lane&15
// ---------------------------------------------------------------------------
__global__ void sage_gemm_kernel(const float* __restrict__ feat,
                                 const float* __restrict__ agg,
                                 const float* __restrict__ rdeg,
                                 const float* __restrict__ Wl,
                                 const float* __restrict__ bl,
                                 const float* __restrict__ Wr,
                                 float* __restrict__ out,
                                 int nTiles, int relu) {
    __shared__ float wlds[128][132];   // Wl, row-padded
    __shared__ float wrds[128][132];   // Wr, row-padded
    __shared__ float hs[16][132];      // feature tile
    __shared__ float ms[16][132];      // mean (agg*rdeg) tile

    const int t     = threadIdx.x;
    const int wave  = t >> 5;
    const int lane  = t & 31;
    const int mrow  = lane & 15;          // A-matrix row / B-matrix col
    const int khalf = (lane >> 4) << 1;   // 0 or 2
    const int o0    = wave * 16;          // output column base for this wave

    // Stage both weight matrices into LDS once (coalesced float4 loads).
    for (int i = t; i < 128 * 32; i += 256) {
        int r = i >> 5;           // 0..127
        int c = (i & 31) * 4;     // 0..124
        *(float4*)&wlds[r][c] = *(const float4*)(Wl + (size_t)r * 128 + c);
        *(float4*)&wrds[r][c] = *(const float4*)(Wr + (size_t)r * 128 + c);
    }
    const float bias = bl[o0 + mrow];
    const int   brow = o0 + mrow;         // this lane's B row in LDS

    for (int tile = blockIdx.x; tile < nTiles; tile += gridDim.x) {
        const int row0 = tile * 16;
        __syncthreads();   // waves done reading LDS tiles from previous iter
        {
            // cooperative load: 16 rows x 128 feats, 8 floats per thread
            int r = t >> 4;
            int c = (t & 15) * 8;
            const float4* fp = (const float4*)(feat + (size_t)(row0 + r) * 128 + c);
            const float4* ap = (const float4*)(agg  + (size_t)(row0 + r) * 128 + c);
            float rd = rdeg[row0 + r];
            float4 f0 = fp[0], f1 = fp[1];
            float4 a0 = ap[0], a1 = ap[1];
            *(float4*)&hs[r][c]     = f0;
            *(float4*)&hs[r][c + 4] = f1;
            *(float4*)&ms[r][c]     = make_float4(a0.x * rd, a0.y * rd, a0.z * rd, a0.w * rd);
            *(float4*)&ms[r][c + 4] = make_float4(a1.x * rd, a1.y * rd, a1.z * rd, a1.w * rd);

            // prefetch next tile's activation lines (global_prefetch_b8)
            int nt = tile + gridDim.x;
            if (nt < nTiles) {
                __builtin_prefetch(feat + (size_t)(nt * 16 + r) * 128 + c, 0, 1);
                __builtin_prefetch(agg  + (size_t)(nt * 16 + r) * 128 + c, 0, 1);
            }
        }
        __syncthreads();

        v8f acc = {};
#pragma unroll
        for (int kk = 0; kk < 32; ++kk) {
            int kb = kk * 4 + khalf;
            v2f aM = *(const v2f*)&ms[mrow][kb];
            v2f bL = *(const v2f*)&wlds[brow][kb];
            acc = __builtin_amdgcn_wmma_f32_16x16x4_f32(
                false, aM, false, bL, (short)0, acc, false, false);
            v2f aH = *(const v2f*)&hs[mrow][kb];
            v2f bR = *(const v2f*)&wrds[brow][kb];
            acc = __builtin_amdgcn_wmma_f32_16x16x4_f32(
                false, aH, false, bR, (short)0, acc, false, false);
        }

        const int rowAdd = (lane >> 4) * 8;
#pragma unroll
        for (int r = 0; r < 8; ++r) {
            float v = acc[r] + bias;
            if (relu) v = fmaxf(v, 0.0f);
            out[(size_t)(row0 + r + rowAdd) * 128 + o0 + mrow] = v;
        }
    }
}

// ---------------------------------------------------------------------------
// Final FC: out[N,40] = feat[N,128] @ Wfc^T[128,40] + bfc
// 96 threads = 3 waves covering 48 columns; columns >= 40 masked.
// ---------------------------------------------------------------------------
__global__ void sage_fc_kernel(const float* __restrict__ feat,
                               const float* __restrict__ W,
                               const float* __restrict__ b,
                               float* __restrict__ out, int ncls) {
    __shared__ float hs[16][132];
    const int t    = threadIdx.x;         // 0..95
    const int wave = t >> 5;              // 0..2
    const int lane = t & 31;
    const int row0 = blockIdx.x * 16;

    for (int i = t; i < 16 * 128; i += 96) {
        int r = i >> 7, c = i & 127;
        hs[r][c] = feat[(size_t)(row0 + r) * 128 + c];
    }
    __syncthreads();

    const int mrow  = lane & 15;
    const int khalf = (lane >> 4) << 1;
    const int col   = wave * 16 + mrow;
    const bool valid = col < ncls;

    v8f acc = {};
#pragma unroll
    for (int kk = 0; kk < 32; ++kk) {
        int kb = kk * 4 + khalf;
        v2f a  = *(const v2f*)&hs[mrow][kb];
        v2f bw = valid ? *(const v2f*)(W + (size_t)col * 128 + kb) : (v2f){0.0f, 0.0f};
        acc = __builtin_amdgcn_wmma_f32_16x16x4_f32(
            false, a, false, bw, (short)0, acc, false, false);
    }

    const float bias = valid ? b[col] : 0.0f;
    const int rowAdd = (lane >> 4) * 8;
#pragma unroll
    for (int r = 0; r < 8; ++r) {
        if (valid)
            out[(size_t)(row0 + r + rowAdd) * ncls + col] = acc[r] + bias;
    }
}

// ---------------------------------------------------------------------------
extern "C" void kernel_launch(void* const* d_in, const int* in_sizes, int n_in,
                              void* d_out, int out_size, void* d_ws, size_t ws_size,
                              hipStream_t stream) {
    const float*     x   = (const float*)d_in[0];
    const long long* ei  = (const long long*)d_in[1];
    const float* Wl0 = (const float*)d_in[2];
    const float* bl0 = (const float*)d_in[3];
    const float* Wr0 = (const float*)d_in[4];
    const float* Wl1 = (const float*)d_in[5];
    const float* bl1 = (const float*)d_in[6];
    const float* Wr1 = (const float*)d_in[7];
    const float* Wl2 = (const float*)d_in[8];
    const float* bl2 = (const float*)d_in[9];
    const float* Wr2 = (const float*)d_in[10];
    const float* Wfc = (const float*)d_in[11];
    const float* bfc = (const float*)d_in[12];

    const int N = in_sizes[0] / 128;      // 100000
    const int E = in_sizes[1] / 2;        // 1600000
    const int NCLS = 40;
    const int nTiles = N / 16;            // 6250 (exact)

    float* ws  = (float*)d_ws;
    float* deg = ws;                                  // N (becomes rdeg)
    float* agg = ws + N;                              // N*128
    float* hb0 = agg + (size_t)N * 128;               // N*128
    float* hb1 = hb0 + (size_t)N * 128;               // N*128

    const long long* src = ei;
    const long long* dst = ei + E;

    float* outp = (float*)d_out;

    // Degrees once (identical for every layer)
    hipMemsetAsync(deg, 0, (size_t)N * sizeof(float), stream);
    sage_deg_kernel<<<(E + 255) / 256, 256, 0, stream>>>(dst, deg, E);
    sage_rdeg_kernel<<<(N + 255) / 256, 256, 0, stream>>>(deg, N);

    const long long aggThreads = (long long)E * 32;
    const int aggBlocks = (int)((aggThreads + 255) / 256);
    const int gemmBlocks = 1024;   // persistent; ~6 row tiles per block

    struct Layer { const float* f; const float* Wl; const float* bl;
                   const float* Wr; float* o; int relu; };
    Layer L[3] = {
        { x,   Wl0, bl0, Wr0, hb0, 1 },
        { hb0, Wl1, bl1, Wr1, hb1, 1 },
        { hb1, Wl2, bl2, Wr2, hb0, 0 },
    };

    for (int l = 0; l < 3; ++l) {
        hipMemsetAsync(agg, 0, (size_t)N * 128 * sizeof(float), stream);
        sage_agg_kernel<<<aggBlocks, 256, 0, stream>>>(src, dst, L[l].f, agg, E);
        sage_gemm_kernel<<<gemmBlocks, 256, 0, stream>>>(
            L[l].f, agg, deg, L[l].Wl, L[l].bl, L[l].Wr, L[l].o, nTiles, L[l].relu);
    }

    sage_fc_kernel<<<nTiles, 96, 0, stream>>>(hb0, Wfc, bfc, outp, NCLS);
}